// StgcnLayer_26929444946466
// MI455X (gfx1250) — compile-verified
//
#include <hip/hip_runtime.h>

// ---------------------------------------------------------------------------
// ST-GCN layer for MI455X (gfx1250), fp32 end-to-end using V_WMMA_F32_16X16X4_F32.
//
// Kernel 1: fused 1x1-conv GEMM + partitioned graph conv -> zsum[n,c,t,v] (d_ws)
// Kernel 2: 9-tap causal temporal sum + BN(eval) + ReLU + residual + ReLU
// ---------------------------------------------------------------------------

typedef float v2f __attribute__((ext_vector_type(2)));
typedef float v8f __attribute__((ext_vector_type(8)));

#define N_    8
#define CIN_  64
#define COUT_ 64
#define P_    3
#define L_    2000
#define V_    25
#define K_    9

#define TT_   8            // timesteps per workgroup
#define VP_   28           // padded V (K of stage-B WMMA: 25 -> 28)
#define COLP_ (TT_ * VP_)  // 224 padded columns of the stage-A GEMM

// LDS strides (floats), padded against bank conflicts
#define XSTR_ 66           // lx[colp][XSTR_]  (cin index 0..63)
#define YSTR_ 226          // ly[c][YSTR_]     (colp index 0..223)
#define WSTR_ 66           // lw[c][WSTR_]     (cin index 0..63)
#define ASTR_ 28           // lAt[v][ASTR_]    (w index 0..27)

// LDS partition offsets in floats
#define OFF_X_ 0
#define OFF_Y_ (OFF_X_ + COLP_ * XSTR_)    // 14784
#define OFF_W_ (OFF_Y_ + COUT_ * YSTR_)    // +14464
#define OFF_A_ (OFF_W_ + COUT_ * WSTR_)    // +4224
#define LDS_FLOATS_ (OFF_A_ + 32 * ASTR_)  // +896 = 34368 floats = 137472 B

__global__ __launch_bounds__(256)
void stgcn_gemm(const float* __restrict__ x,
                const float* __restrict__ A,
                const float* __restrict__ cw,
                float* __restrict__ zsum)
{
    extern __shared__ float smem[];
    float* lx  = smem + OFF_X_;   // x tile, column-major: lx[colp*XSTR_ + cin]
    float* ly  = smem + OFF_Y_;   // Yp tile: ly[c*YSTR_ + colp]
    float* lw  = smem + OFF_W_;   // weight slice: lw[c*WSTR_ + cin]
    float* lAt = smem + OFF_A_;   // transposed adjacency: lAt[v*ASTR_ + w]

    const int tid  = threadIdx.x;
    const int lane = tid & 31;
    const int wid  = tid >> 5;     // wave id 0..7
    const int half = lane >> 4;    // 0: lanes 0-15, 1: lanes 16-31
    const int l16  = lane & 15;

    const int n  = blockIdx.y;
    const int t0 = blockIdx.x * TT_;    // 250 blocks cover L exactly

    // ---------------- phase 0: stage x tile into LDS (zero-pad w=25..27) ----
    for (int i = tid; i < CIN_ * COLP_; i += 256) {
        int k    = i / COLP_;          // cin
        int colp = i - k * COLP_;      // padded (t,w) column
        int t    = colp / VP_;
        int w    = colp - t * VP_;
        float val = 0.0f;
        if (w < V_)
            val = x[((n * CIN_ + k) * L_ + (t0 + t)) * V_ + w];
        lx[colp * XSTR_ + k] = val;
    }

    // per-wave persistent accumulators: wave owns (mt,vt), j indexes t 0..7
    const int mtw = wid >> 1;          // 0..3  -> c tile
    const int vtw = wid & 1;           // 0..1  -> v tile
    v8f accB[TT_];
    const v8f vzero = {0.f, 0.f, 0.f, 0.f, 0.f, 0.f, 0.f, 0.f};
#pragma unroll
    for (int j = 0; j < TT_; ++j) accB[j] = vzero;

    for (int p = 0; p < P_; ++p) {
        __syncthreads();  // previous iteration's readers of lw/lAt/ly done

        // stage weight slice W_p (64x64) and A_p^T (zero-padded to 32x28)
        for (int i = tid; i < COUT_ * CIN_; i += 256) {
            int m = i >> 6;
            int k = i & 63;
            lw[m * WSTR_ + k] = cw[(p * COUT_ + m) * CIN_ + k];
        }
        for (int i = tid; i < 32 * ASTR_; i += 256) {
            int v = i / ASTR_;
            int w = i - v * ASTR_;
            float val = 0.0f;
            if (v < V_ && w < V_)
                val = A[(p * V_ + w) * V_ + v];   // lAt[v][w] = A[p][w][v]
            lAt[i] = val;
        }
        __syncthreads();

        // ------------- stage A: Yp(64 x 224) = W_p(64x64) @ X(64x224) -------
        // A-operand lane layout (16x4 f32): lanes 0-15 -> (K,K+1), 16-31 -> (K+2,K+3)
        for (int tile = wid; tile < 4 * 14; tile += 8) {
            int mt = tile / 14, nt = tile - mt * 14;
            int m0 = mt * 16,  n0 = nt * 16;
            const float* ap = &lw[(m0 + l16) * WSTR_ + 2 * half];
            const float* bp = &lx[(n0 + l16) * XSTR_ + 2 * half];
            v8f acc = vzero;
#pragma unroll
            for (int k = 0; k < CIN_; k += 4) {
                v2f av = *(const v2f*)(ap + k);
                v2f bv = *(const v2f*)(bp + k);
                acc = __builtin_amdgcn_wmma_f32_16x16x4_f32(
                          false, av, false, bv, (short)0, acc, false, false);
            }
            // C layout: VGPR i -> row m0+i (lanes 0-15) / m0+i+8 (lanes 16-31)
#pragma unroll
            for (int i = 0; i < 8; ++i)
                ly[(m0 + i + 8 * half) * YSTR_ + n0 + l16] = acc[i];
        }
        __syncthreads();

        // ------------- stage B: accB[t] += Yp_t(64x25) @ A_p(25x25) ---------
#pragma unroll
        for (int j = 0; j < TT_; ++j) {
            const float* ap = &ly[(mtw * 16 + l16) * YSTR_ + j * VP_ + 2 * half];
            const float* bp = &lAt[(vtw * 16 + l16) * ASTR_ + 2 * half];
            v8f acc = accB[j];
#pragma unroll
            for (int k = 0; k < VP_; k += 4) {
                v2f av = *(const v2f*)(ap + k);
                v2f bv = *(const v2f*)(bp + k);
                acc = __builtin_amdgcn_wmma_f32_16x16x4_f32(
                          false, av, false, bv, (short)0, acc, false, false);
            }
            accB[j] = acc;
        }
    }

    // ---------------- write zsum[n,c,t,v] -----------------------------------
    const int v = vtw * 16 + l16;
    if (v < V_) {
#pragma unroll
        for (int j = 0; j < TT_; ++j) {
#pragma unroll
            for (int i = 0; i < 8; ++i) {
                int c = mtw * 16 + i + 8 * half;
                zsum[((n * COUT_ + c) * L_ + (t0 + j)) * V_ + v] = accB[j][i];
            }
        }
    }
}

// ---------------------------------------------------------------------------
// Kernel 2: causal 9-tap temporal sum + BN(eval) + ReLU + residual + ReLU
// ---------------------------------------------------------------------------
__global__ __launch_bounds__(256)
void stgcn_post(const float* __restrict__ zsum,
                const float* __restrict__ x,
                const float* __restrict__ gamma,
                const float* __restrict__ beta,
                const float* __restrict__ mean,
                const float* __restrict__ var,
                float* __restrict__ out)
{
    long idx = (long)blockIdx.x * blockDim.x + threadIdx.x;
    const long total = (long)N_ * COUT_ * L_ * V_;
    if (idx >= total) return;

    int  v  = (int)(idx % V_);
    long r  = idx / V_;
    int  l  = (int)(r % L_);
    long nc = r / L_;
    int  c  = (int)(nc % COUT_);

    const float* zp = zsum + nc * (long)L_ * V_ + v;
    int tmin = l - (K_ - 1);
    if (tmin < 0) tmin = 0;
    float s = 0.0f;
    for (int t = tmin; t <= l; ++t)
        s += zp[(long)t * V_];

    float sc = gamma[c] * rsqrtf(var[c] + 1e-5f);
    float sh = beta[c] - mean[c] * sc;
    float bn = s * sc + sh;
    bn = bn > 0.0f ? bn : 0.0f;
    float res = bn + x[idx];
    out[idx] = res > 0.0f ? res : 0.0f;
}

extern "C" void kernel_launch(void* const* d_in, const int* in_sizes, int n_in,
                              void* d_out, int out_size, void* d_ws, size_t ws_size,
                              hipStream_t stream) {
    const float* x     = (const float*)d_in[0];
    const float* A     = (const float*)d_in[1];
    const float* cw    = (const float*)d_in[2];
    const float* gamma = (const float*)d_in[3];
    const float* beta  = (const float*)d_in[4];
    const float* mean  = (const float*)d_in[5];
    const float* var   = (const float*)d_in[6];
    float* out  = (float*)d_out;
    float* zsum = (float*)d_ws;   // needs N*COUT*L*V*4 = 102.4 MB of scratch

    dim3 g1(L_ / TT_, N_);        // 250 x 8 workgroups
    size_t shbytes = (size_t)LDS_FLOATS_ * sizeof(float);  // 137472 B
    stgcn_gemm<<<g1, 256, shbytes, stream>>>(x, A, cw, zsum);

    long total = (long)N_ * COUT_ * L_ * V_;
    int blocks = (int)((total + 255) / 256);
    stgcn_post<<<blocks, 256, 0, stream>>>(zsum, x, gamma, beta, mean, var, out);
}